// GraphNetWithAttentionPooling_79594333929610
// MI455X (gfx1250) — compile-verified
//
#include <hip/hip_runtime.h>
#include <hip/hip_bf16.h>
#include <math.h>

typedef __attribute__((ext_vector_type(2))) float v2f;
typedef __attribute__((ext_vector_type(8))) float v8f;

#define N_NODES 100000
#define N_EDGES 1600000
#define K_KEEP  80000          // ceil(0.8 * N)
#define F_IN    128
#define PCH     16             // per-node channels layer1
#define HEADS   4
#define HID     128
#define NEG_SLOPE 0.2f
#define BIG_NEG  -1.0e9f

// ---------- helpers ----------
__device__ __forceinline__ unsigned fkey(float f) {      // order-preserving float->uint
  unsigned u = __float_as_uint(f);
  return (u & 0x80000000u) ? ~u : (u | 0x80000000u);
}
__device__ __forceinline__ float kval(unsigned k) {
  return (k & 0x80000000u) ? __uint_as_float(k & 0x7FFFFFFFu) : __uint_as_float(~k);
}
__device__ __forceinline__ float lrelu(float x) { return x > 0.f ? x : NEG_SLOPE * x; }

// ---------- small init / constants ----------
__global__ void k_init_small(unsigned* st, float* attacc, unsigned* mxk) {
  if (blockIdx.x) return;
  int t = threadIdx.x;
  if (t == 0) {
    st[0] = 0u;                       // radix prefix
    st[1] = (unsigned)K_KEEP;         // remaining rank
    st[2] = 0u;                       // selected counter
    st[3] = 0u;                       // tie counter
    st[4] = fkey(-3.0e38f);           // gate max key
    ((float*)st)[5] = 0.f;            // gate exp-sum
  }
  if (t < HID) { attacc[t] = 0.f; mxk[t] = fkey(-3.0e38f); }
}

__global__ void k_ce(const float* We1, const float* ae1,
                     const float* We2, const float* ae2, float* stf) {
  if (threadIdx.x == 0 && blockIdx.x == 0) {
    for (int hd = 0; hd < HEADS; ++hd) {
      float s = 0.f;
      for (int j = 0; j < 4; ++j) s += We1[hd * 4 + j] * ae1[hd * 4 + j];
      stf[8 + hd] = s;
    }
    float s2 = 0.f;
    for (int c = 0; c < HID; ++c) s2 += We2[c] * ae2[c];
    stf[12] = s2;
  }
}

// ---------- layer 1: h = x @ W1 via V_WMMA_F32_16X16X4_F32 ----------
__global__ void k_gemm1(const float* __restrict__ x, const float* __restrict__ W1,
                        float* __restrict__ h) {
  int lane = threadIdx.x & 31;
  int wid  = blockIdx.x * (blockDim.x >> 5) + (threadIdx.x >> 5);
  if (wid >= N_NODES / 16) return;
  int col  = lane & 15;
  int hi   = lane >> 4;       // lanes 0-15: K={0,1}, lanes 16-31: K={2,3}
  int koff = hi * 2;
  const float* xr = x + (size_t)(wid * 16 + col) * F_IN;
  v8f c = {0.f, 0.f, 0.f, 0.f, 0.f, 0.f, 0.f, 0.f};
#pragma unroll 4
  for (int k0 = 0; k0 < F_IN; k0 += 4) {
    v2f a, b;
    a[0] = xr[k0 + koff];
    a[1] = xr[k0 + koff + 1];
    b[0] = W1[(k0 + koff) * PCH + col];
    b[1] = W1[(k0 + koff + 1) * PCH + col];
    c = __builtin_amdgcn_wmma_f32_16x16x4_f32(false, a, false, b, (short)0, c, false, false);
  }
  int mbase = wid * 16 + hi * 8;
#pragma unroll
  for (int i = 0; i < 8; ++i) h[(size_t)(mbase + i) * PCH + col] = c[i];
}

__global__ void k_alpha1(const float* __restrict__ h, const float* __restrict__ as1,
                         const float* __restrict__ ad1,
                         float* __restrict__ als, float* __restrict__ ald) {
  int n = blockIdx.x * blockDim.x + threadIdx.x;
  if (n >= N_NODES) return;
  const float* hr = h + (size_t)n * PCH;
#pragma unroll
  for (int hd = 0; hd < HEADS; ++hd) {
    float s = 0.f, d = 0.f;
#pragma unroll
    for (int j = 0; j < 4; ++j) {
      float v = hr[hd * 4 + j];
      s += v * as1[hd * 4 + j];
      d += v * ad1[hd * 4 + j];
    }
    als[n * HEADS + hd] = s;
    ald[n * HEADS + hd] = d;
  }
}

__global__ void k_init1(unsigned* m1, float* den1, float* h1acc) {
  int g = blockIdx.x * blockDim.x + threadIdx.x;
  if (g >= N_NODES * PCH) return;
  h1acc[g] = 0.f;
  if (g < N_NODES * HEADS) { m1[g] = fkey(BIG_NEG); den1[g] = 0.f; }
}

__global__ void k_elog1(const int* __restrict__ src, const int* __restrict__ dst,
                        const float* __restrict__ ea,
                        const float* __restrict__ als, const float* __restrict__ ald,
                        const float* __restrict__ stf, float* __restrict__ lg,
                        unsigned* __restrict__ m1) {
  int e = blockIdx.x * blockDim.x + threadIdx.x;
  if (e >= N_EDGES) return;
  int s = src[e], d = dst[e];
  float a = ea[e];
#pragma unroll
  for (int hd = 0; hd < HEADS; ++hd) {
    float l = lrelu(als[s * HEADS + hd] + ald[d * HEADS + hd] + a * stf[8 + hd]);
    lg[(size_t)e * HEADS + hd] = l;
    atomicMax(&m1[d * HEADS + hd], fkey(l));
  }
}

__global__ void k_enum1(const int* __restrict__ dst, float* __restrict__ lg,
                        const unsigned* __restrict__ m1, float* __restrict__ den1) {
  int e = blockIdx.x * blockDim.x + threadIdx.x;
  if (e >= N_EDGES) return;
  int d = dst[e];
#pragma unroll
  for (int hd = 0; hd < HEADS; ++hd) {
    float num = __expf(lg[(size_t)e * HEADS + hd] - kval(m1[d * HEADS + hd]));
    lg[(size_t)e * HEADS + hd] = num;
    atomicAdd(&den1[d * HEADS + hd], num);
  }
}

__global__ void k_eagg1(const int* __restrict__ src, const int* __restrict__ dst,
                        const float* __restrict__ lg, const float* __restrict__ den1,
                        const float* __restrict__ h, float* __restrict__ h1acc) {
  int g = blockIdx.x * blockDim.x + threadIdx.x;   // E*16 threads
  int e = g >> 4, c = g & 15;
  if (e >= N_EDGES) return;
  int s = src[e], d = dst[e], hd = c >> 2;
  float coef = lg[(size_t)e * HEADS + hd] / fmaxf(den1[d * HEADS + hd], 1e-16f);
  atomicAdd(&h1acc[(size_t)d * PCH + c], coef * h[(size_t)s * PCH + c]);
}

__global__ void k_nfin1(float* __restrict__ h1, const float* __restrict__ b1,
                        const float* __restrict__ Wrel, const float* __restrict__ brel,
                        const float* __restrict__ Wroot,
                        float* __restrict__ p, float* __restrict__ score) {
  int n = blockIdx.x * blockDim.x + threadIdx.x;
  if (n >= N_NODES) return;
  float pv = 0.f, rv = 0.f;
  float* hr = h1 + (size_t)n * PCH;
#pragma unroll
  for (int c = 0; c < PCH; ++c) {
    float v = fmaxf(hr[c] + b1[c], 0.f);   // relu(gat1 + b1)
    hr[c] = v;
    pv += v * Wrel[c];
    rv += v * Wroot[c];
  }
  p[n] = pv;
  score[n] = rv + brel[0];
}

__global__ void k_escore(const int* __restrict__ src, const int* __restrict__ dst,
                         const float* __restrict__ p, float* __restrict__ score) {
  int e = blockIdx.x * blockDim.x + threadIdx.x;
  if (e >= N_EDGES) return;
  atomicAdd(&score[dst[e]], p[src[e]]);
}

// ---------- top-k via 4-round MSD radix select ----------
__global__ void k_keys(const float* __restrict__ score, unsigned* __restrict__ key) {
  int n = blockIdx.x * blockDim.x + threadIdx.x;
  if (n < N_NODES) key[n] = fkey(score[n]);
}
__global__ void k_zhist(unsigned* hist) {
  if (blockIdx.x == 0 && threadIdx.x < 256) hist[threadIdx.x] = 0u;
}
__global__ void k_hist(const unsigned* __restrict__ key, const unsigned* __restrict__ st,
                       unsigned* __restrict__ hist, int round) {
  int n = blockIdx.x * blockDim.x + threadIdx.x;
  if (n >= N_NODES) return;
  unsigned u = key[n];
  if (round == 0 || (u >> (32 - 8 * round)) == st[0])
    atomicAdd(&hist[(u >> (24 - 8 * round)) & 255u], 1u);
}
__global__ void k_pick(unsigned* st, const unsigned* hist) {
  if (threadIdx.x || blockIdx.x) return;
  unsigned prefix = st[0];
  int remaining = (int)st[1];
  int acc = 0, bin = 0;
  for (int b = 255; b >= 0; --b) {
    int c = (int)hist[b];
    if (acc + c >= remaining) { bin = b; break; }
    acc += c;
  }
  st[0] = (prefix << 8) | (unsigned)bin;
  st[1] = (unsigned)(remaining - acc);
}

__global__ void k_select(const unsigned* __restrict__ key, const float* __restrict__ score,
                         const float* __restrict__ h1, unsigned* st,
                         int* __restrict__ newidx, float* __restrict__ x2) {
  int n = blockIdx.x * blockDim.x + threadIdx.x;
  if (n >= N_NODES) return;
  unsigned T = st[0];
  int take_eq = (int)st[1];
  unsigned u = key[n];
  int my = -1;
  if (u > T) my = atomicAdd((int*)&st[2], 1);
  else if (u == T) {
    int t = atomicAdd((int*)&st[3], 1);
    if (t < take_eq) my = atomicAdd((int*)&st[2], 1);
  }
  newidx[n] = my;
  if (my >= 0) {
    float tv = tanhf(score[n]);
    const float* hr = h1 + (size_t)n * PCH;
    float* xr = x2 + (size_t)my * PCH;
#pragma unroll
    for (int c = 0; c < PCH; ++c) xr[c] = hr[c] * tv;
  }
}

// ---------- layer 2: h2pre = x2 @ W2 via WMMA ----------
__global__ void k_gemm2(const float* __restrict__ x2, const float* __restrict__ W2,
                        float* __restrict__ h2pre) {
  int lane = threadIdx.x & 31;
  int wid  = blockIdx.x * (blockDim.x >> 5) + (threadIdx.x >> 5);
  if (wid >= (K_KEEP / 16) * (HID / 16)) return;
  int mt = wid >> 3, nt = wid & 7;
  int cl = lane & 15, hi = lane >> 4, koff = hi * 2;
  int gcol = nt * 16 + cl;
  const float* xr = x2 + (size_t)(mt * 16 + cl) * PCH;
  v8f c = {0.f, 0.f, 0.f, 0.f, 0.f, 0.f, 0.f, 0.f};
#pragma unroll
  for (int k0 = 0; k0 < PCH; k0 += 4) {
    v2f a, b;
    a[0] = xr[k0 + koff];
    a[1] = xr[k0 + koff + 1];
    b[0] = W2[(k0 + koff) * HID + gcol];
    b[1] = W2[(k0 + koff + 1) * HID + gcol];
    c = __builtin_amdgcn_wmma_f32_16x16x4_f32(false, a, false, b, (short)0, c, false, false);
  }
  int mbase = mt * 16 + hi * 8;
#pragma unroll
  for (int i = 0; i < 8; ++i) h2pre[(size_t)(mbase + i) * HID + gcol] = c[i];
}

__global__ void k_alpha2(const float* __restrict__ h2pre, const float* __restrict__ as2,
                         const float* __restrict__ ad2,
                         float* __restrict__ als2, float* __restrict__ ald2) {
  int j = blockIdx.x * blockDim.x + threadIdx.x;
  if (j >= K_KEEP) return;
  const float* hr = h2pre + (size_t)j * HID;
  float s = 0.f, d = 0.f;
  for (int c = 0; c < HID; ++c) { float v = hr[c]; s += v * as2[c]; d += v * ad2[c]; }
  als2[j] = s; ald2[j] = d;
}

__global__ void k_init2(unsigned* m2, float* den2, float* out2) {
  int g = blockIdx.x * blockDim.x + threadIdx.x;
  if (g >= K_KEEP * HID) return;
  out2[g] = 0.f;
  if (g < K_KEEP) { m2[g] = fkey(BIG_NEG); den2[g] = 0.f; }
}

__global__ void k_elog2(const int* __restrict__ src, const int* __restrict__ dst,
                        const int* __restrict__ newidx, const float* __restrict__ ea,
                        const float* __restrict__ als2, const float* __restrict__ ald2,
                        const float* __restrict__ stf,
                        float* __restrict__ num2, unsigned* __restrict__ m2) {
  int e = blockIdx.x * blockDim.x + threadIdx.x;
  if (e >= N_EDGES) return;
  int sn = newidx[src[e]], dn = newidx[dst[e]];
  if (sn >= 0 && dn >= 0) {
    float l = lrelu(als2[sn] + ald2[dn] + ea[e] * stf[12]);
    num2[e] = l;
    atomicMax(&m2[dn], fkey(l));
  }
}

__global__ void k_enum2(const int* __restrict__ src, const int* __restrict__ dst,
                        const int* __restrict__ newidx,
                        float* __restrict__ num2, const unsigned* __restrict__ m2,
                        float* __restrict__ den2) {
  int e = blockIdx.x * blockDim.x + threadIdx.x;
  if (e >= N_EDGES) return;
  int sn = newidx[src[e]], dn = newidx[dst[e]];
  if (sn >= 0 && dn >= 0) {
    float num = __expf(num2[e] - kval(m2[dn]));
    num2[e] = num;
    atomicAdd(&den2[dn], num);
  } else {
    num2[e] = 0.f;   // masked edges contribute nothing (matches reference num=0)
  }
}

__global__ void k_eagg2(const int* __restrict__ src, const int* __restrict__ dst,
                        const int* __restrict__ newidx,
                        const float* __restrict__ num2, const float* __restrict__ den2,
                        const float* __restrict__ h2pre, float* __restrict__ out2) {
  int g = blockIdx.x * blockDim.x + threadIdx.x;   // E * 32 threads, one wave per edge
  int e = g >> 5, lane = g & 31;
  if (e >= N_EDGES) return;
  int sn = newidx[src[e]], dn = newidx[dst[e]];
  if (sn < 0 || dn < 0) return;
  float coef = num2[e] / fmaxf(den2[dn], 1e-16f);
  const float* hp = h2pre + (size_t)sn * HID;
  float* op = out2 + (size_t)dn * HID;
#pragma unroll
  for (int i = 0; i < 4; ++i) {
    int c = lane + 32 * i;
    atomicAdd(&op[c], coef * hp[c]);
  }
}

__global__ void k_nfin2(float* __restrict__ out2, const float* __restrict__ b2,
                        const float* __restrict__ Wg, const float* __restrict__ bg,
                        float* __restrict__ gbuf, unsigned* st) {
  int j = blockIdx.x * blockDim.x + threadIdx.x;
  if (j >= K_KEEP) return;
  float* hr = out2 + (size_t)j * HID;
  float g = 0.f;
  for (int c = 0; c < HID; ++c) {
    float v = fmaxf(hr[c] + b2[c], 0.f);   // h2 = relu(gat2 + b2)
    hr[c] = v;
    g += v * Wg[c];
  }
  g += bg[0];
  gbuf[j] = g;
  atomicMax(&st[4], fkey(g));
}

__global__ void k_gsum(const float* __restrict__ gbuf, unsigned* st) {
  int j = blockIdx.x * blockDim.x + threadIdx.x;
  if (j >= K_KEEP) return;
  atomicAdd(((float*)st) + 5, __expf(gbuf[j] - kval(st[4])));
}

__global__ void k_pool(const float* __restrict__ out2, const float* __restrict__ gbuf,
                       const unsigned* st, float* __restrict__ attacc,
                       unsigned* __restrict__ mxk) {
  int g = blockIdx.x * blockDim.x + threadIdx.x;   // K * 128
  if (g >= K_KEEP * HID) return;
  int j = g >> 7, c = g & 127;
  float w = __expf(gbuf[j] - kval(st[4])) / ((const float*)st)[5];
  float v = out2[g];
  atomicAdd(&attacc[c], w * v);
  atomicMax(&mxk[c], fkey(v));
}

__global__ void k_out(const float* __restrict__ attacc, const unsigned* __restrict__ mxk,
                      float* __restrict__ out) {
  int c = threadIdx.x;
  if (blockIdx.x == 0 && c < HID) {
    out[c] = attacc[c];
    out[HID + c] = kval(mxk[c]);
  }
}

// ---------- launcher ----------
extern "C" void kernel_launch(void* const* d_in, const int* in_sizes, int n_in,
                              void* d_out, int out_size, void* d_ws, size_t ws_size,
                              hipStream_t stream) {
  const float* x     = (const float*)d_in[0];
  const int*   eidx  = (const int*)d_in[1];
  const float* eattr = (const float*)d_in[2];
  const float* W1    = (const float*)d_in[3];
  const float* as1   = (const float*)d_in[4];
  const float* ad1   = (const float*)d_in[5];
  const float* We1   = (const float*)d_in[6];
  const float* ae1   = (const float*)d_in[7];
  const float* b1    = (const float*)d_in[8];
  const float* Wrel  = (const float*)d_in[9];
  const float* brel  = (const float*)d_in[10];
  const float* Wroot = (const float*)d_in[11];
  const float* W2    = (const float*)d_in[12];
  const float* as2   = (const float*)d_in[13];
  const float* ad2   = (const float*)d_in[14];
  const float* We2   = (const float*)d_in[15];
  const float* ae2   = (const float*)d_in[16];
  const float* b2    = (const float*)d_in[17];
  const float* Wg    = (const float*)d_in[18];
  const float* bg    = (const float*)d_in[19];
  float* out = (float*)d_out;
  const int* src = eidx;
  const int* dst = eidx + N_EDGES;

  // workspace layout (units: 4-byte words)
  float* W = (float*)d_ws;
  constexpr size_t O_H    = 0;
  constexpr size_t O_ALS1 = O_H    + (size_t)N_NODES * PCH;
  constexpr size_t O_ALD1 = O_ALS1 + (size_t)N_NODES * HEADS;
  constexpr size_t O_LG1  = O_ALD1 + (size_t)N_NODES * HEADS;
  constexpr size_t O_M1   = O_LG1  + (size_t)N_EDGES * HEADS;
  constexpr size_t O_DEN1 = O_M1   + (size_t)N_NODES * HEADS;
  constexpr size_t O_H1   = O_DEN1 + (size_t)N_NODES * HEADS;
  constexpr size_t O_SC   = O_H1   + (size_t)N_NODES * PCH;
  constexpr size_t O_P    = O_SC   + (size_t)N_NODES;
  constexpr size_t O_KEY  = O_P    + (size_t)N_NODES;
  constexpr size_t O_NIX  = O_KEY  + (size_t)N_NODES;
  constexpr size_t O_X2   = O_NIX  + (size_t)N_NODES;
  constexpr size_t O_H2P  = O_X2   + (size_t)K_KEEP * PCH;
  constexpr size_t O_ALS2 = O_H2P  + (size_t)K_KEEP * HID;
  constexpr size_t O_ALD2 = O_ALS2 + (size_t)K_KEEP;
  constexpr size_t O_M2   = O_ALD2 + (size_t)K_KEEP;
  constexpr size_t O_DEN2 = O_M2   + (size_t)K_KEEP;
  constexpr size_t O_NUM2 = O_DEN2 + (size_t)K_KEEP;
  constexpr size_t O_OUT2 = O_NUM2 + (size_t)N_EDGES;
  constexpr size_t O_G    = O_OUT2 + (size_t)K_KEEP * HID;
  constexpr size_t O_ATT  = O_G    + (size_t)K_KEEP;
  constexpr size_t O_MXK  = O_ATT  + (size_t)HID;
  constexpr size_t O_HIST = O_MXK  + (size_t)HID;
  constexpr size_t O_ST   = O_HIST + 256;

  float*    h     = W + O_H;
  float*    als1  = W + O_ALS1;
  float*    ald1  = W + O_ALD1;
  float*    lg1   = W + O_LG1;
  unsigned* m1    = (unsigned*)(W + O_M1);
  float*    den1  = W + O_DEN1;
  float*    h1    = W + O_H1;
  float*    score = W + O_SC;
  float*    p     = W + O_P;
  unsigned* key   = (unsigned*)(W + O_KEY);
  int*      nix   = (int*)(W + O_NIX);
  float*    x2    = W + O_X2;
  float*    h2p   = W + O_H2P;
  float*    als2  = W + O_ALS2;
  float*    ald2  = W + O_ALD2;
  unsigned* m2    = (unsigned*)(W + O_M2);
  float*    den2  = W + O_DEN2;
  float*    num2  = W + O_NUM2;
  float*    out2  = W + O_OUT2;
  float*    gbuf  = W + O_G;
  float*    attac = W + O_ATT;
  unsigned* mxk   = (unsigned*)(W + O_MXK);
  unsigned* hist  = (unsigned*)(W + O_HIST);
  unsigned* st    = (unsigned*)(W + O_ST);
  float*    stf   = (float*)st;

  const int B = 256;
  auto G = [](long n) { return (unsigned)((n + 255) / 256); };

  k_init_small<<<1, B, 0, stream>>>(st, attac, mxk);
  k_ce<<<1, 64, 0, stream>>>(We1, ae1, We2, ae2, stf);

  // layer 1
  k_gemm1<<<G((long)(N_NODES / 16) * 32), B, 0, stream>>>(x, W1, h);
  k_alpha1<<<G(N_NODES), B, 0, stream>>>(h, as1, ad1, als1, ald1);
  k_init1<<<G((long)N_NODES * PCH), B, 0, stream>>>(m1, den1, h1);
  k_elog1<<<G(N_EDGES), B, 0, stream>>>(src, dst, eattr, als1, ald1, stf, lg1, m1);
  k_enum1<<<G(N_EDGES), B, 0, stream>>>(dst, lg1, m1, den1);
  k_eagg1<<<G((long)N_EDGES * 16), B, 0, stream>>>(src, dst, lg1, den1, h, h1);
  k_nfin1<<<G(N_NODES), B, 0, stream>>>(h1, b1, Wrel, brel, Wroot, p, score);

  // SAGPool score + radix-select top-k
  k_escore<<<G(N_EDGES), B, 0, stream>>>(src, dst, p, score);
  k_keys<<<G(N_NODES), B, 0, stream>>>(score, key);
  for (int r = 0; r < 4; ++r) {
    k_zhist<<<1, B, 0, stream>>>(hist);
    k_hist<<<G(N_NODES), B, 0, stream>>>(key, st, hist, r);
    k_pick<<<1, 1, 0, stream>>>(st, hist);
  }
  k_select<<<G(N_NODES), B, 0, stream>>>(key, score, h1, st, nix, x2);

  // layer 2
  k_gemm2<<<G((long)(K_KEEP / 16) * (HID / 16) * 32), B, 0, stream>>>(x2, W2, h2p);
  k_alpha2<<<G(K_KEEP), B, 0, stream>>>(h2p, as2, ad2, als2, ald2);
  k_init2<<<G((long)K_KEEP * HID), B, 0, stream>>>(m2, den2, out2);
  k_elog2<<<G(N_EDGES), B, 0, stream>>>(src, dst, nix, eattr, als2, ald2, stf, num2, m2);
  k_enum2<<<G(N_EDGES), B, 0, stream>>>(src, dst, nix, num2, m2, den2);
  k_eagg2<<<G((long)N_EDGES * 32), B, 0, stream>>>(src, dst, nix, num2, den2, h2p, out2);
  k_nfin2<<<G(K_KEEP), B, 0, stream>>>(out2, b2, Wg, bg, gbuf, st);

  // attentional + max aggregation
  k_gsum<<<G(K_KEEP), B, 0, stream>>>(gbuf, st);
  k_pool<<<G((long)K_KEEP * HID), B, 0, stream>>>(out2, gbuf, st, attac, mxk);
  k_out<<<1, B, 0, stream>>>(attac, mxk, out);
}